// Decoder_79405355369049
// MI455X (gfx1250) — compile-verified
//
#include <hip/hip_runtime.h>
#include <hip/hip_bf16.h>
#include <math.h>

#define VOCAB 32000
#define EMB   256
#define HID   512
#define SP1   513           // SEQ + 1
#define G4    (4*HID)       // 2048 gate rows

typedef __attribute__((ext_vector_type(2))) float v2f;
typedef __attribute__((ext_vector_type(8))) float v8f;

// ------------------------------------------------------------------
// Embedding gather: tokens = [SOS=1, input_seq...] -> emb [513][256]
// ------------------------------------------------------------------
__global__ void gather_emb(const int* __restrict__ seq,
                           const float* __restrict__ embedding,
                           float* __restrict__ emb_out) {
    int t = blockIdx.x;            // 0..512
    int e = threadIdx.x;           // 0..255
    int tok = (t == 0) ? 1 : seq[t - 1];
    emb_out[t * EMB + e] = embedding[(size_t)tok * EMB + e];
}

// ------------------------------------------------------------------
// Permute W_hh[2048][512] -> Wp[k][c], c = 4*u + g  (g in i,f,g,o order)
// so LSTM thread j reads a coalesced float2 at column 2j for each k.
// ------------------------------------------------------------------
__global__ void permute_whh(const float* __restrict__ whh, float* __restrict__ wp) {
    int gid = blockIdx.x * blockDim.x + threadIdx.x;
    if (gid >= HID * G4) return;
    int k = gid / G4;
    int c = gid % G4;
    int u = c >> 2, g = c & 3;
    wp[gid] = whh[((size_t)(g * HID + u)) * HID + k];
}

// ------------------------------------------------------------------
// WMMA f32 GEMM:  C[m,n] = sum_k A[m*K+k] * B[n*K+k] + bias1[n] (+bias2[n])
// A:[M,K] row-major, B:[N,K] row-major (i.e. C = A * B^T).
// Block = 256 threads = 8 waves; tile = 16(M) x 128(N); K chunk = 32.
// V_WMMA_F32_16X16X4_F32, fragments prefetched so ds loads clause and
// the 8 WMMAs issue back-to-back on the accumulator chain.
// LDS row stride 36: bank-conflict-free for 16-lane column reads
// (gcd(36,64)=4 -> 36m mod 64 distinct for m=0..15) and 16B-aligned.
// ------------------------------------------------------------------
#define BN 128
#define BK 32
#define LDP 36
__global__ void __launch_bounds__(256)
gemm_tn_wmma(const float* __restrict__ A, const float* __restrict__ B,
             const float* __restrict__ bias1, const float* __restrict__ bias2,
             float* __restrict__ C, int M, int N, int K) {
    __shared__ float As[16][LDP];
    __shared__ float Bs[BN][LDP];

    const int tid   = threadIdx.x;
    const int wave  = tid >> 5;        // 0..7
    const int lane  = tid & 31;
    const int lhalf = lane >> 4;       // 0 or 1
    const int frow  = lane & 15;       // fragment row (M for A, N for B)
    const int m0    = blockIdx.y * 16;
    const int n0    = blockIdx.x * BN;

    v8f acc = {};

    for (int k0 = 0; k0 < K; k0 += BK) {
        // Stage A tile (16 x 32) as float4: 128 vec loads, threads 0..127
        if (tid < 128) {
            int m  = tid >> 3;            // 0..15
            int k4 = (tid & 7) * 4;       // 0,4,...,28
            int gm = m0 + m;
            float4 v = make_float4(0.f, 0.f, 0.f, 0.f);
            if (gm < M) v = *(const float4*)(A + (size_t)gm * K + k0 + k4);
            *(float4*)(&As[m][k4]) = v;
        }
        // Stage B tile (128 x 32) as float4: 1024 vec loads, 4 per thread
        #pragma unroll
        for (int i = 0; i < 4; ++i) {
            int idx = tid + i * 256;
            int n  = idx >> 3;            // 0..127
            int k4 = (idx & 7) * 4;
            float4 v = *(const float4*)(B + (size_t)(n0 + n) * K + k0 + k4);
            *(float4*)(&Bs[n][k4]) = v;
        }
        __syncthreads();

        const int nr = wave * 16 + frow;
        // Prefetch all fragments for this K chunk (one dscnt wait total)
        v2f af[8], bf[8];
        #pragma unroll
        for (int kk = 0; kk < 8; ++kk) {
            af[kk].x = As[frow][kk * 4 + lhalf * 2];
            af[kk].y = As[frow][kk * 4 + lhalf * 2 + 1];
            bf[kk].x = Bs[nr][kk * 4 + lhalf * 2];
            bf[kk].y = Bs[nr][kk * 4 + lhalf * 2 + 1];
        }
        #pragma unroll
        for (int kk = 0; kk < 8; ++kk) {
            acc = __builtin_amdgcn_wmma_f32_16x16x4_f32(
                false, af[kk], false, bf[kk], (short)0, acc, false, false);
        }
        __syncthreads();
    }

    const int gn = n0 + wave * 16 + frow;
    float bv = bias1 ? bias1[gn] : 0.0f;
    if (bias2) bv += bias2[gn];
    #pragma unroll
    for (int r = 0; r < 8; ++r) {
        int gm = m0 + lhalf * 8 + r;   // D: M = r + half*8, N = lane&15
        if (gm < M) C[(size_t)gm * N + gn] = acc[r] + bv;
    }
}

// ------------------------------------------------------------------
// Sequential LSTM: one persistent workgroup (1024 thr = 32 wave32s).
// Thread j owns permuted gate rows 2j,2j+1 (u = j>>1; even j -> i,f,
// odd j -> g,o).  h in LDS; Wp loads are coalesced float2 per k.
// Xg already holds W_ih@x_t + b_ih + b_hh in natural [g*512+u] order.
// ------------------------------------------------------------------
__global__ void __launch_bounds__(1024)
lstm_seq(const float* __restrict__ Xg, const float* __restrict__ Wp,
         const float* __restrict__ h0, const float* __restrict__ c0,
         float* __restrict__ hs, float* __restrict__ out_tail) {
    __shared__ float h_s[HID];
    __shared__ float c_s[HID];
    __shared__ float act[4 * HID];

    const int j   = threadIdx.x;        // 0..1023
    const int u   = j >> 1;
    const bool ev = (j & 1) == 0;
    const int g0  = ev ? 0 : 2;
    const int g1  = g0 + 1;

    if (j < HID) { h_s[j] = h0[j]; c_s[j] = c0[j]; }
    __syncthreads();

    for (int t = 0; t < SP1; ++t) {
        float d0 = Xg[(size_t)t * G4 + g0 * HID + u];
        float d1 = Xg[(size_t)t * G4 + g1 * HID + u];
        const float* wrow = Wp + 2 * j;
        #pragma unroll 4
        for (int k = 0; k < HID; ++k) {
            float2 w = *(const float2*)(wrow + (size_t)k * G4);
            float hk = h_s[k];
            d0 = fmaf(w.x, hk, d0);
            d1 = fmaf(w.y, hk, d1);
        }
        if (ev) {
            act[0 * HID + u] = 1.0f / (1.0f + expf(-d0));   // i
            act[1 * HID + u] = 1.0f / (1.0f + expf(-d1));   // f
        } else {
            act[2 * HID + u] = tanhf(d0);                   // g
            act[3 * HID + u] = 1.0f / (1.0f + expf(-d1));   // o
        }
        __syncthreads();
        if (j < HID) {
            float iv = act[j], fv = act[HID + j];
            float gv = act[2 * HID + j], ov = act[3 * HID + j];
            float c  = fmaf(fv, c_s[j], iv * gv);
            c_s[j] = c;
            float hn = ov * tanhf(c);
            h_s[j] = hn;
            hs[(size_t)t * HID + j] = hn;
        }
        __syncthreads();
    }

    if (j < HID) {
        out_tail[j]       = h_s[j];   // hT
        out_tail[HID + j] = c_s[j];   // cT
    }
}

// ------------------------------------------------------------------
extern "C" void kernel_launch(void* const* d_in, const int* in_sizes, int n_in,
                              void* d_out, int out_size, void* d_ws, size_t ws_size,
                              hipStream_t stream) {
    const int*   seq   = (const int*)  d_in[0];   // [1,512]
    const float* h0    = (const float*)d_in[1];   // [1,1,512]
    const float* c0    = (const float*)d_in[2];   // [1,1,512]
    const float* emb_w = (const float*)d_in[3];   // [32000,256]
    const float* W_ih  = (const float*)d_in[4];   // [2048,256]
    const float* W_hh  = (const float*)d_in[5];   // [2048,512]
    const float* b_ih  = (const float*)d_in[6];   // [2048]
    const float* b_hh  = (const float*)d_in[7];   // [2048]
    const float* W_out = (const float*)d_in[8];   // [32000,512]
    const float* b_out = (const float*)d_in[9];   // [32000]

    float* out = (float*)d_out;                   // logits[513*32000] + hT[512] + cT[512]
    float* ws  = (float*)d_ws;

    // Workspace layout (floats)
    float* emb = ws;                                   // 513*256   = 131328
    float* Xg  = emb + SP1 * EMB;                      // 513*2048  = 1050624
    float* Wp  = Xg  + SP1 * G4;                       // 512*2048  = 1048576
    float* hs  = Wp  + HID * G4;                       // 513*512   = 262656

    const size_t LOGITS = (size_t)SP1 * VOCAB;

    // 1. Gather embeddings for [SOS, input_seq]
    gather_emb<<<SP1, EMB, 0, stream>>>(seq, emb_w, emb);

    // 2. Permute W_hh for coalesced recurrent matvec
    permute_whh<<<(HID * G4 + 255) / 256, 256, 0, stream>>>(W_hh, Wp);

    // 3. Input projection for all steps: Xg = emb @ W_ih^T + b_ih + b_hh
    gemm_tn_wmma<<<dim3(G4 / BN, (SP1 + 15) / 16), 256, 0, stream>>>(
        emb, W_ih, b_ih, b_hh, Xg, SP1, G4, EMB);

    // 4. Sequential LSTM over 513 steps (single WGP, persistent)
    lstm_seq<<<1, 1024, 0, stream>>>(Xg, Wp, h0, c0, hs, out + LOGITS);

    // 5. Output projection: logits = hs @ W_out^T + b_out  (memory-bound)
    gemm_tn_wmma<<<dim3(VOCAB / BN, (SP1 + 15) / 16), 256, 0, stream>>>(
        hs, W_out, b_out, nullptr, out, SP1, VOCAB, HID);
}